// ComboLoss_18622978196097
// MI455X (gfx1250) — compile-verified
//
#include <hip/hip_runtime.h>
#include <hip/hip_bf16.h>
#include <stdint.h>

#define B_TOTAL   65536
#define NMODES    5
#define TSTEPS    50
#define ROW       505          // NMODES*TSTEPS*2 + NMODES
#define GT_ROW    100
#define TILE      32           // samples per wave (one TDM tile)
#define NUM_TILES (B_TOTAL / TILE)   // 2048

typedef __attribute__((ext_vector_type(4))) unsigned int v4u;
typedef __attribute__((ext_vector_type(8))) int v8i;
typedef __attribute__((ext_vector_type(4))) int v4i;

__device__ __forceinline__ unsigned rotl32(unsigned v, unsigned r) {
  return (v << r) | (v >> (32u - r));
}

// Exact replication of jax.random.randint(jax.random.key(42), (65536,), 0, 5)
// threefry-2x32, key = (0, 42), counts = arange(65536) split into two halves.
__device__ unsigned threefry_bits(unsigned i) {
  const unsigned k0 = 0u, k1 = 42u;
  const unsigned k2 = k0 ^ k1 ^ 0x1BD11BDAu;
  const unsigned half = (i >= 32768u) ? 1u : 0u;
  unsigned x0 = half ? (i - 32768u) : i;
  unsigned x1 = x0 + 32768u;
  x0 += k0; x1 += k1;
  const unsigned ra[4] = {13u, 15u, 26u, 6u};
  const unsigned rb[4] = {17u, 29u, 16u, 24u};
#pragma unroll
  for (int j = 0; j < 4; ++j) { x0 += x1; x1 = rotl32(x1, ra[j]); x1 ^= x0; }
  x0 += k1; x1 += k2 + 1u;
#pragma unroll
  for (int j = 0; j < 4; ++j) { x0 += x1; x1 = rotl32(x1, rb[j]); x1 ^= x0; }
  x0 += k2; x1 += k0 + 2u;
#pragma unroll
  for (int j = 0; j < 4; ++j) { x0 += x1; x1 = rotl32(x1, ra[j]); x1 ^= x0; }
  x0 += k0; x1 += k1 + 3u;
#pragma unroll
  for (int j = 0; j < 4; ++j) { x0 += x1; x1 = rotl32(x1, rb[j]); x1 ^= x0; }
  x0 += k1; x1 += k2 + 4u;
#pragma unroll
  for (int j = 0; j < 4; ++j) { x0 += x1; x1 = rotl32(x1, ra[j]); x1 ^= x0; }
  x0 += k2; x1 += k0 + 5u;
  return half ? x1 : x0;
}

extern "C" __global__ void __launch_bounds__(32)
combo_loss_main(const float* __restrict__ path_pred,
                const float* __restrict__ path_gt,
                const float* __restrict__ cr_pred,
                const float* __restrict__ cr_gt,
                float* __restrict__ partials) {
  extern __shared__ float ppbuf[];       // TILE * ROW floats (64,640 B)
  const int lane = threadIdx.x;          // wave32: 0..31
  const int tile = blockIdx.x;           // 0..NUM_TILES-1
  const int s    = tile * TILE + lane;   // sample id

  // ---------- TDM: DMA a 32x505 f32 tile (row stride 505) into LDS ----------
  const uint64_t ga = (uint64_t)(uintptr_t)(path_pred + (size_t)tile * TILE * ROW);
  const unsigned lds_addr = (unsigned)(uintptr_t)(&ppbuf[0]);

  // D# group 0: count=1 valid, lds_addr, 57-bit global_addr, type=2 (bits 127:126)
  v4u g0 = { 1u,
             lds_addr,
             (unsigned)(ga & 0xFFFFFFFFu),
             ((unsigned)((ga >> 32) & 0x01FFFFFFu)) | 0x80000000u };
  // D# group 1: data_size=4B (bits 17:16 = 2); tensor_dim0=505; tensor_dim1=32;
  //             tile_dim0=505; tile_dim1=32; tile_dim2=0; dim0_stride=505
  v8i g1 = { (int)0x00020000u,           // workgroup_mask=0, data_size=2 (4B)
             (int)((unsigned)ROW << 16),  // tensor_dim0 low16 @ bits 63:48
             (int)((unsigned)TILE << 16), // tensor_dim1 low16 @ bits 95:80
             (int)((unsigned)ROW << 16),  // tile_dim0 @ bits 127:112
             (int)TILE,                   // tile_dim1 @ [15:0]; tile_dim2=0
             (int)ROW,                    // tensor_dim0_stride low32
             0, 0 };
  v4i gz4 = { 0, 0, 0, 0 };              // groups 2/3 unused (2D tile)
  v8i gz8 = { 0, 0, 0, 0, 0, 0, 0, 0 };
  __builtin_amdgcn_tensor_load_to_lds(g0, g1, gz4, gz4, gz8, 0);
  __builtin_amdgcn_s_wait_tensorcnt(0);  // per-wave: only this wave reads its LDS

  const float* __restrict__ row = ppbuf + lane * ROW;  // 505 % 64 = 57 -> conflict-free
  const float2* __restrict__ gt2 =
      reinterpret_cast<const float2*>(path_gt + (size_t)s * GT_ROW);  // 400B rows, 16B aligned

  // ---------- per-mode mean distance ----------
  float dist[NMODES] = {0.f, 0.f, 0.f, 0.f, 0.f};
  for (int t = 0; t < TSTEPS; ++t) {
    const float2 g = gt2[t];
#pragma unroll
    for (int m = 0; m < NMODES; ++m) {
      const float dx = row[m * 100 + 2 * t]     - g.x;
      const float dy = row[m * 100 + 2 * t + 1] - g.y;
      dist[m] += sqrtf(dx * dx + dy * dy);
    }
  }

  // ---------- angle gate + first-min argmin ----------
  const float2 rl  = gt2[TSTEPS - 1];
  const float refx = rl.x, refy = rl.y;
  const float nref = sqrtf(refx * refx + refy * refy);
  const bool refnan = (refx != refx) || (refy != refy);

  int best = -1;
  float bestd = __builtin_inff();
#pragma unroll
  for (int m = 0; m < NMODES; ++m) {
    const float tx = row[m * 100 + 98];
    const float ty = row[m * 100 + 99];
    const float np = nref * sqrtf(tx * tx + ty * ty);
    const float dt = refx * tx + refy * ty;
    const float denom = (np == 0.f) ? 1.f : np;
    const float c = fminf(1.f, fmaxf(-1.f, dt / denom));
    float ang = (np == 0.f) ? 0.f : acosf(c) * 57.29577951308232f;  // degrees
    if (refnan || (tx != tx) || (ty != ty)) ang = 180.f - 1e-5f;
    const float dm = dist[m] * (1.f / TSTEPS);
    if ((ang <= 5.0f) && (dm < bestd)) { bestd = dm; best = m; }
  }
  if (best < 0) {  // no eligible mode -> JAX random fallback
    const unsigned bits = threefry_bits((unsigned)s);
    best = (int)(((bits >> 16) + (bits & 0xFFFFu)) % 5u);
  }

  // ---------- smooth-L1 on best mode ----------
  const float* __restrict__ br = row + best * 100;
  float reg = 0.f;
  for (int t = 0; t < TSTEPS; ++t) {
    const float2 g = gt2[t];
    const float d0 = br[2 * t]     - g.x;
    const float d1 = br[2 * t + 1] - g.y;
    const float a0 = fabsf(d0), a1 = fabsf(d1);
    reg += (a0 < 1.f) ? 0.5f * d0 * d0 : (a0 - 0.5f);
    reg += (a1 < 1.f) ? 0.5f * d1 * d1 : (a1 - 0.5f);
  }
  reg *= (1.f / (TSTEPS * 2));

  // ---------- cross entropy over mode logits ----------
  float mp[NMODES];
#pragma unroll
  for (int m = 0; m < NMODES; ++m) mp[m] = row[NMODES * TSTEPS * 2 + m];
  float mx = mp[0];
#pragma unroll
  for (int m = 1; m < NMODES; ++m) mx = fmaxf(mx, mp[m]);
  float se = 0.f;
#pragma unroll
  for (int m = 0; m < NMODES; ++m) se += expf(mp[m] - mx);
  const float ce = -(mp[best] - mx - logf(se));

  // ---------- BCE term for this sample ----------
  const float p = cr_pred[s];
  const float y = cr_gt[s];
  const float lp  = fmaxf(logf(p), -100.f);
  const float l1p = fmaxf(log1pf(-p), -100.f);
  const float bce = -(y * lp + (1.f - y) * l1p);

  // ---------- deterministic wave reduction ----------
  float v = ce + reg + bce;
#pragma unroll
  for (int off = 16; off > 0; off >>= 1) v += __shfl_xor(v, off, 32);
  if (lane == 0) partials[tile] = v;
}

extern "C" __global__ void __launch_bounds__(256)
combo_loss_reduce(const float* __restrict__ partials, float* __restrict__ out) {
  __shared__ float wsum[8];
  float v = 0.f;
  for (int i = threadIdx.x; i < NUM_TILES; i += 256) v += partials[i];
#pragma unroll
  for (int off = 16; off > 0; off >>= 1) v += __shfl_xor(v, off, 32);
  if ((threadIdx.x & 31) == 0) wsum[threadIdx.x >> 5] = v;
  __syncthreads();
  if (threadIdx.x == 0) {
    float t = 0.f;
#pragma unroll
    for (int w = 0; w < 8; ++w) t += wsum[w];
    out[0] = t * (1.0f / B_TOTAL);
  }
}

extern "C" void kernel_launch(void* const* d_in, const int* in_sizes, int n_in,
                              void* d_out, int out_size, void* d_ws, size_t ws_size,
                              hipStream_t stream) {
  const float* path_pred = (const float*)d_in[0];
  const float* path_gt   = (const float*)d_in[1];
  const float* cr_pred   = (const float*)d_in[2];
  const float* cr_gt     = (const float*)d_in[3];
  // d_in[4] = log_vars (unused by the reference output)
  float* partials = (float*)d_ws;        // NUM_TILES floats = 8 KB
  float* out      = (float*)d_out;

  const size_t shmem = (size_t)TILE * ROW * sizeof(float);  // 64,640 B per block
  combo_loss_main<<<NUM_TILES, 32, shmem, stream>>>(path_pred, path_gt, cr_pred,
                                                    cr_gt, partials);
  combo_loss_reduce<<<1, 256, 0, stream>>>(partials, out);
}